// SelfAttention_51324859187587
// MI455X (gfx1250) — compile-verified
//
#include <hip/hip_runtime.h>

#define DIM   1024
#define BATCH 8
#define SEQ   2048

typedef __attribute__((ext_vector_type(2))) float v2f;
typedef __attribute__((ext_vector_type(8))) float v8f;

// ---------------------------------------------------------------------------
// Stage A1: partial column sums of x over 128-row chunks of the sequence.
// grid (DIM/256, 16 chunks, B), block 256.  partial[(b*16+chunk)*DIM + d]
// ---------------------------------------------------------------------------
__global__ void colsum_partial_k(const float* __restrict__ x,
                                 float* __restrict__ partial) {
    const int d     = blockIdx.x * 256 + threadIdx.x;
    const int chunk = blockIdx.y;
    const int b     = blockIdx.z;
    const float* p = x + ((size_t)b * SEQ + (size_t)chunk * 128) * DIM + d;
    float s = 0.f;
#pragma unroll 8
    for (int i = 0; i < 128; ++i) s += p[(size_t)i * DIM];
    partial[(size_t)(b * 16 + chunk) * DIM + d] = s;
}

// ---------------------------------------------------------------------------
// Stage A2: reduce 16 partials -> Xs[16][DIM] (rows 0..7 = batch sums,
// rows 8..15 zero padding for the M=16 WMMA tile).  grid (DIM/256, 16).
// ---------------------------------------------------------------------------
__global__ void colsum_final_k(const float* __restrict__ partial,
                               float* __restrict__ Xs) {
    const int d = blockIdx.x * 256 + threadIdx.x;
    const int m = blockIdx.y;   // 0..15
    float s = 0.f;
    if (m < BATCH) {
#pragma unroll
        for (int c = 0; c < 16; ++c)
            s += partial[(size_t)(m * 16 + c) * DIM + d];
    }
    Xs[(size_t)m * DIM + d] = s;
}

// ---------------------------------------------------------------------------
// WMMA GEMM:  D(16 x DIM) = A(16 x DIM) @ Brows^T (+ bias), fp32 WMMA.
// Brows is row-major [DIM_out][DIM] (torch Linear weight), contraction along
// its fast dim, so both A and B fragments are contiguous float2 per lane.
// One wave per 16-wide N tile: grid(8) x block(256) = 64 waves = 64 tiles.
//   A frag (16x4):  lane L: row = L&15, K = k0 + 2*(L>>4) + {0,1}
//   B frag (4x16):  lane L: col = tile*16 + (L&15), K same as A
//   C/D:            vgpr r: M = r + 8*(L>>4), N = tile*16 + (L&15)
// ---------------------------------------------------------------------------
__global__ void gemm_wmma_k(const float* __restrict__ A,
                            const float* __restrict__ Brows,
                            const float* __restrict__ bias,
                            float* __restrict__ D) {
    const int lane  = threadIdx.x & 31;
    const int ntile = blockIdx.x * 8 + (threadIdx.x >> 5);   // 0..63
    const int g     = lane >> 4;                              // half-wave group
    const int mh    = lane & 15;
    const int n     = ntile * 16 + mh;

    const float* arow = A     + (size_t)mh * DIM;
    const float* brow = Brows + (size_t)n  * DIM;

    v8f c = {};
    for (int k0 = 0; k0 < DIM; k0 += 4) {
        v2f a = *(const v2f*)(arow + k0 + 2 * g);
        v2f b = *(const v2f*)(brow + k0 + 2 * g);
        // D = A x B + C, fp32, 16x16x4
        c = __builtin_amdgcn_wmma_f32_16x16x4_f32(
                /*neg_a=*/false, a, /*neg_b=*/false, b,
                /*c_mod=*/(short)0, c, /*reuse_a=*/false, /*reuse_b=*/false);
    }

    const float bv = bias ? bias[n] : 0.f;
#pragma unroll
    for (int r = 0; r < 8; ++r) {
        const int mr = r + 8 * g;
        D[(size_t)mr * DIM + n] = c[r] + bv;
    }
}

// ---------------------------------------------------------------------------
// Stage C: out[row,:] = LayerNorm(x[row,:] + y2[b,:]) * gamma + beta
// One wave per row (32 lanes x 8 float4 = 1024 elems kept in registers).
// grid(2048) x block(256) -> 8 rows per block, 16384 rows total.
// ---------------------------------------------------------------------------
__global__ void resid_ln_k(const float* __restrict__ x,
                           const float* __restrict__ y2,
                           const float* __restrict__ gamma,
                           const float* __restrict__ beta,
                           float* __restrict__ out) {
    const int lane = threadIdx.x & 31;
    const int row  = blockIdx.x * 8 + (threadIdx.x >> 5);    // 0..16383
    const int b    = row >> 11;                               // SEQ = 2048 rows/batch

    const float4* xr = (const float4*)(x  + (size_t)row * DIM);
    const float4* yr = (const float4*)(y2 + (size_t)b   * DIM);

    float4 h[8];
    float s = 0.f, ss = 0.f;
#pragma unroll
    for (int i = 0; i < 8; ++i) {
        const int idx = lane + 32 * i;
        float4 hv = xr[idx];
        const float4 yv = yr[idx];
        hv.x += yv.x; hv.y += yv.y; hv.z += yv.z; hv.w += yv.w;
        h[i] = hv;
        s  += (hv.x + hv.y) + (hv.z + hv.w);
        ss += hv.x * hv.x + hv.y * hv.y + hv.z * hv.z + hv.w * hv.w;
    }
#pragma unroll
    for (int off = 16; off >= 1; off >>= 1) {
        s  += __shfl_xor(s,  off, 32);
        ss += __shfl_xor(ss, off, 32);
    }
    const float mean = s * (1.0f / DIM);
    const float var  = fmaxf(ss * (1.0f / DIM) - mean * mean, 0.0f);
    const float rstd = rsqrtf(var + 1e-5f);

    float4* outr = (float4*)(out + (size_t)row * DIM);
    const float4* g4 = (const float4*)gamma;
    const float4* b4 = (const float4*)beta;
#pragma unroll
    for (int i = 0; i < 8; ++i) {
        const int idx = lane + 32 * i;
        const float4 g = g4[idx], bb = b4[idx], hv = h[i];
        float4 o;
        o.x = (hv.x - mean) * rstd * g.x + bb.x;
        o.y = (hv.y - mean) * rstd * g.y + bb.y;
        o.z = (hv.z - mean) * rstd * g.z + bb.z;
        o.w = (hv.w - mean) * rstd * g.w + bb.w;
        outr[idx] = o;
    }
}

extern "C" void kernel_launch(void* const* d_in, const int* in_sizes, int n_in,
                              void* d_out, int out_size, void* d_ws, size_t ws_size,
                              hipStream_t stream) {
    const float* x     = (const float*)d_in[0];   // [8, 2048, 1024]
    const float* w_qkv = (const float*)d_in[1];   // [3072, 1024]
    const float* w_fc  = (const float*)d_in[2];   // [1024, 1024]
    const float* b_fc  = (const float*)d_in[3];   // [1024]
    const float* gamma = (const float*)d_in[4];   // [1024]
    const float* beta  = (const float*)d_in[5];   // [1024]
    float* out = (float*)d_out;

    // Small persistent scratch in d_ws (192 KB): Xs, Vsum, y2 (each 16 x 1024 f32)
    float* Xs   = (float*)d_ws;
    float* Vsum = Xs   + 16 * DIM;
    float* y2   = Vsum + 16 * DIM;
    // Large transient scratch (512 KB of partial column sums) lives in d_out:
    // consumed by colsum_final_k before resid_ln_k overwrites all of d_out.
    float* partial = out;

    // 1) xsum[b,:] = sum_s x[b,s,:]
    colsum_partial_k<<<dim3(DIM / 256, 16, BATCH), 256, 0, stream>>>(x, partial);
    colsum_final_k  <<<dim3(DIM / 256, 16),        256, 0, stream>>>(partial, Xs);

    // 2) Vsum = xsum @ W_v^T   (W_v = rows [2048,3072) of w_qkv)
    gemm_wmma_k<<<dim3(8), 256, 0, stream>>>(Xs, w_qkv + (size_t)2 * DIM * DIM,
                                             nullptr, Vsum);
    // 3) y2 = Vsum @ W_fc^T + b_fc
    gemm_wmma_k<<<dim3(8), 256, 0, stream>>>(Vsum, w_fc, b_fc, y2);

    // 4) out = LN(x + broadcast(y2)) * gamma + beta
    resid_ln_k<<<dim3(SEQ * BATCH / 8), 256, 0, stream>>>(x, y2, gamma, beta, out);
}